// MoleGNN_66099546685676
// MI455X (gfx1250) — compile-verified
//
#include <hip/hip_runtime.h>

typedef __attribute__((ext_vector_type(2))) float v2f;
typedef __attribute__((ext_vector_type(8))) float v8f;

#define HDIM 32
#define NEG_SLOPE 0.2f

// ---------- helpers: order-preserving float<->uint key for atomicMax ----------
__device__ __forceinline__ unsigned fkey(float f) {
    unsigned b = __float_as_uint(f);
    return (b & 0x80000000u) ? ~b : (b | 0x80000000u);
}
__device__ __forceinline__ float keyf(unsigned k) {
    unsigned b = (k & 0x80000000u) ? (k & 0x7FFFFFFFu) : ~k;
    return __uint_as_float(b);
}

// ---------- GEMM: H[n x 32] = X[n x K] @ W[K x 32] via V_WMMA_F32_16X16X4_F32 ----------
// One wave32 computes one 16x16 tile of H. 8 waves / 256-thread block.
__global__ void gat_gemm_f32wmma(const float* __restrict__ X,
                                 const float* __restrict__ W,
                                 float* __restrict__ H,
                                 int n_rows, int K) {
    const int gwave = (blockIdx.x * blockDim.x + threadIdx.x) >> 5;
    const int lane  = threadIdx.x & 31;
    const int row_blk = gwave >> 1;          // 16-row block
    const int col_blk = gwave & 1;           // two 16-col blocks (HDIM=32)
    if (row_blk * 16 + 16 > n_rows) return;  // wave-uniform guard (EXEC stays all-1s)

    const int mrow  = row_blk * 16 + (lane & 15);
    const int khalf = (lane >> 4) * 2;       // lanes 0-15: K={k,k+1}; 16-31: K={k+2,k+3}
    const int ncol  = col_blk * 16 + (lane & 15);

    v8f c = {};
    for (int k = 0; k < K; k += 4) {
        // A tile (16x4 f32): per-lane float2 from row mrow, cols k+khalf..k+khalf+1
        const float* ap = X + (size_t)mrow * K + k + khalf;
        v2f a; a.x = ap[0]; a.y = ap[1];
        // B tile (4x16 f32): VGPR0 = row k+khalf, VGPR1 = row k+khalf+1 (col ncol)
        v2f b;
        b.x = W[(size_t)(k + khalf) * HDIM + ncol];
        b.y = W[(size_t)(k + khalf + 1) * HDIM + ncol];
        c = __builtin_amdgcn_wmma_f32_16x16x4_f32(false, a, false, b,
                                                  (short)0, c, false, false);
    }
    // D layout: VGPR v -> lanes 0-15 row (blk+v), lanes 16-31 row (blk+v+8); N = lane&15
    const int base_row = row_blk * 16 + (lane >> 4) * 8;
#pragma unroll
    for (int v = 0; v < 8; ++v)
        H[(size_t)(base_row + v) * HDIM + ncol] = c[v];
}

// ---------- per-node attention logits: es = h.a_s, ed = h.a_d ----------
__global__ void gat_logits(const float* __restrict__ H,
                           const float* __restrict__ a_s,
                           const float* __restrict__ a_d,
                           float* __restrict__ es, float* __restrict__ ed, int n) {
    int i = blockIdx.x * blockDim.x + threadIdx.x;
    if (i >= n) return;
    const float* hp = H + (size_t)i * HDIM;
    float s = 0.f, d = 0.f;
#pragma unroll
    for (int f = 0; f < HDIM; ++f) { float v = hp[f]; s += v * a_s[f]; d += v * a_d[f]; }
    es[i] = s; ed[i] = d;
}

// ---------- zero accumulators for one layer ----------
__global__ void gat_init(float* __restrict__ acc, float* __restrict__ denom,
                         unsigned* __restrict__ mkey, int n) {
    int idx = blockIdx.x * blockDim.x + threadIdx.x;
    if (idx < n * HDIM) acc[idx] = 0.f;
    if (idx < n) { denom[idx] = 0.f; mkey[idx] = 0u; }  // key 0 < key(any finite float)
}

// ---------- edge pass 1: segment max of leaky_relu(es[src]+ed[dst]) per dst ----------
__global__ void gat_edge_max(const int* __restrict__ ei,
                             const float* __restrict__ es, const float* __restrict__ ed,
                             unsigned* __restrict__ mkey, int E, int n) {
    int t = blockIdx.x * blockDim.x + threadIdx.x;
    if (t >= E + n) return;
    int s, d;
    if (t < E) { s = ei[t]; d = ei[E + t]; } else { s = d = t - E; }  // self loops
    float e = es[s] + ed[d];
    e = e > 0.f ? e : NEG_SLOPE * e;
    atomicMax(&mkey[d], fkey(e));
}

// ---------- edge pass 2: ex = exp(e - m[dst]); denom += ex; acc[dst] += ex*h[src] ----------
__global__ void gat_edge_acc(const int* __restrict__ ei,
                             const float* __restrict__ es, const float* __restrict__ ed,
                             const unsigned* __restrict__ mkey,
                             float* __restrict__ denom,
                             const float* __restrict__ H, float* __restrict__ acc,
                             int E, int n) {
    int t = blockIdx.x * blockDim.x + threadIdx.x;
    if (t >= E + n) return;
    int s, d;
    if (t < E) { s = ei[t]; d = ei[E + t]; } else { s = d = t - E; }
    float e = es[s] + ed[d];
    e = e > 0.f ? e : NEG_SLOPE * e;
    float ex = __expf(e - keyf(mkey[d]));
    atomicAdd(&denom[d], ex);
    const float4* hp = (const float4*)(H + (size_t)s * HDIM);
    float* ap = acc + (size_t)d * HDIM;
#pragma unroll
    for (int q = 0; q < HDIM / 4; ++q) {
        float4 hv = hp[q];
        atomicAdd(ap + q * 4 + 0, ex * hv.x);
        atomicAdd(ap + q * 4 + 1, ex * hv.y);
        atomicAdd(ap + q * 4 + 2, ex * hv.z);
        atomicAdd(ap + q * 4 + 3, ex * hv.w);
    }
}

// ---------- normalize + bias -> next layer input ----------
__global__ void gat_finalize(const float* __restrict__ acc, const float* __restrict__ denom,
                             const float* __restrict__ b, float* __restrict__ xn, int n) {
    int idx = blockIdx.x * blockDim.x + threadIdx.x;
    if (idx >= n * HDIM) return;
    int i = idx >> 5, f = idx & 31;
    xn[idx] = acc[idx] / denom[i] + b[f];
}

// ---------- mean pool ----------
__global__ void pool_zero(float* __restrict__ out, float* __restrict__ cnt, int g32, int g) {
    int idx = blockIdx.x * blockDim.x + threadIdx.x;
    if (idx < g32) out[idx] = 0.f;
    if (idx < g) cnt[idx] = 0.f;
}
__global__ void pool_acc(const float* __restrict__ x, const int* __restrict__ batch,
                         float* __restrict__ out, float* __restrict__ cnt, int n) {
    int idx = blockIdx.x * blockDim.x + threadIdx.x;
    if (idx >= n * HDIM) return;
    int i = idx >> 5, f = idx & 31;
    int g = batch[i];
    atomicAdd(&out[(size_t)g * HDIM + f], x[idx]);
    if (f == 0) atomicAdd(&cnt[g], 1.0f);
}
__global__ void pool_div(float* __restrict__ out, const float* __restrict__ cnt, int g32) {
    int idx = blockIdx.x * blockDim.x + threadIdx.x;
    if (idx >= g32) return;
    out[idx] /= fmaxf(cnt[idx >> 5], 1.0f);
}

extern "C" void kernel_launch(void* const* d_in, const int* in_sizes, int n_in,
                              void* d_out, int out_size, void* d_ws, size_t ws_size,
                              hipStream_t stream) {
    const float* x0    = (const float*)d_in[0];
    const int*   ei    = (const int*)d_in[1];
    const int*   batch = (const int*)d_in[2];
    const int E   = in_sizes[1] / 2;
    const int N   = in_sizes[2];
    const int G32 = out_size;            // G * HDIM
    const int G   = G32 / HDIM;

    const float* Wl[4] = {(const float*)d_in[3],  (const float*)d_in[7],
                          (const float*)d_in[11], (const float*)d_in[15]};
    const float* Sl[4] = {(const float*)d_in[4],  (const float*)d_in[8],
                          (const float*)d_in[12], (const float*)d_in[16]};
    const float* Dl[4] = {(const float*)d_in[5],  (const float*)d_in[9],
                          (const float*)d_in[13], (const float*)d_in[17]};
    const float* Bl[4] = {(const float*)d_in[6],  (const float*)d_in[10],
                          (const float*)d_in[14], (const float*)d_in[18]};
    const int Kl[4] = {128, HDIM, HDIM, HDIM};

    // workspace carve-out (256B aligned)
    char* w = (char*)d_ws;
    auto carve = [&](size_t bytes) { void* p = (void*)w; w += (bytes + 255) & ~(size_t)255; return p; };
    float*    hbuf  = (float*)carve((size_t)N * HDIM * 4);
    float*    acc   = (float*)carve((size_t)N * HDIM * 4);
    float*    xbuf  = (float*)carve((size_t)N * HDIM * 4);
    float*    es    = (float*)carve((size_t)N * 4);
    float*    ed    = (float*)carve((size_t)N * 4);
    float*    denom = (float*)carve((size_t)N * 4);
    unsigned* mkey  = (unsigned*)carve((size_t)N * 4);
    float*    cnt   = (float*)carve((size_t)G * 4);

    const int BT = 256;
    const int Etot = E + N;
    const int gemm_waves = ((N + 15) / 16) * 2;           // one wave per 16x16 tile

    const float* xin = x0;
    for (int l = 0; l < 4; ++l) {
        gat_gemm_f32wmma<<<dim3((gemm_waves * 32 + BT - 1) / BT), dim3(BT), 0, stream>>>(
            xin, Wl[l], hbuf, N, Kl[l]);
        gat_logits<<<dim3((N + BT - 1) / BT), dim3(BT), 0, stream>>>(hbuf, Sl[l], Dl[l], es, ed, N);
        gat_init<<<dim3((N * HDIM + BT - 1) / BT), dim3(BT), 0, stream>>>(acc, denom, mkey, N);
        gat_edge_max<<<dim3((Etot + BT - 1) / BT), dim3(BT), 0, stream>>>(ei, es, ed, mkey, E, N);
        gat_edge_acc<<<dim3((Etot + BT - 1) / BT), dim3(BT), 0, stream>>>(ei, es, ed, mkey, denom,
                                                                          hbuf, acc, E, N);
        gat_finalize<<<dim3((N * HDIM + BT - 1) / BT), dim3(BT), 0, stream>>>(acc, denom, Bl[l], xbuf, N);
        xin = xbuf;
    }

    pool_zero<<<dim3((G32 + BT - 1) / BT), dim3(BT), 0, stream>>>((float*)d_out, cnt, G32, G);
    pool_acc<<<dim3((N * HDIM + BT - 1) / BT), dim3(BT), 0, stream>>>(xbuf, batch, (float*)d_out, cnt, N);
    pool_div<<<dim3((G32 + BT - 1) / BT), dim3(BT), 0, stream>>>((float*)d_out, cnt, G32);
}